// SlicPatchEmbed_3478923510490
// MI455X (gfx1250) — compile-verified
//
#include <hip/hip_runtime.h>
#include <hip/hip_bf16.h>
#include <stdint.h>

#define IMG 224
#define HW (IMG * IMG)
#define PSZ 16
#define NCH 3
#define NSEG 100
#define KMITERS 10
#define ROWS (NSEG * PSZ * NCH)        /* 4800 rows of the stage-1 intermediate */
#define SPF (NSEG * PSZ * PSZ * NCH)   /* 76800 elems per image in sp */

typedef __attribute__((ext_vector_type(2))) float v2f;
typedef __attribute__((ext_vector_type(8))) float v8f;
typedef __attribute__((ext_vector_type(4))) unsigned int u32x4;
typedef __attribute__((ext_vector_type(8))) int i32x8;
typedef __attribute__((ext_vector_type(4))) int i32x4;

#ifdef __has_builtin
#if __has_builtin(__builtin_amdgcn_tensor_load_to_lds) && __has_builtin(__builtin_amdgcn_s_wait_tensorcnt)
#define USE_TDM 1
#endif
#endif

// ---------------------------------------------------------------------------
// Antialiased linear 224->16 resize matrix (row-normalized triangle kernel),
// identical to jax.image.resize(method='linear', antialias=True) on eye(224).
// ---------------------------------------------------------------------------
__global__ void build_A_kernel(float* __restrict__ A) {
  int p = threadIdx.x;
  if (p >= PSZ) return;
  const float scale = 14.0f;                       // src/dst = 224/16
  float center = (p + 0.5f) * scale - 0.5f;        // sample position in source
  float sum = 0.0f;
  for (int h = 0; h < IMG; ++h) {
    float t = fabsf((float)h - center) / scale;    // stretched triangle kernel
    float w = t < 1.0f ? 1.0f - t : 0.0f;
    A[p * IMG + h] = w;
    sum += w;
  }
  float inv = 1.0f / sum;
  for (int h = 0; h < IMG; ++h) A[p * IMG + h] *= inv;
}

// ---------------------------------------------------------------------------
// Simplified SLIC: grid-init k-means in (rgb, compactness-scaled yx) space.
// One workgroup per image; centers + accumulators in LDS (ds_add_f32).
// ---------------------------------------------------------------------------
__global__ __launch_bounds__(1024) void slic_kernel(const float* __restrict__ x,
                                                    int* __restrict__ labels) {
  const int b = blockIdx.x;
  const float* img = x + (size_t)b * NCH * HW;
  const float ratio = 10.0f / 22.4f;               // COMPACT / sqrt(HW/K)
  __shared__ float cen[NSEG][5];
  __shared__ float acc[NSEG][5];
  __shared__ float cnt[NSEG];
  const int tid = threadIdx.x, nthr = blockDim.x;

  if (tid < NSEG) {                                // grid init of centers
    int gi = tid / 10, gj = tid % 10;
    int cy = (int)((gi + 0.5f) * 22.4f);
    int cx = (int)((gj + 0.5f) * 22.4f);
    int pidx = cy * IMG + cx;
    cen[tid][0] = img[pidx];
    cen[tid][1] = img[HW + pidx];
    cen[tid][2] = img[2 * HW + pidx];
    cen[tid][3] = (float)cy * ratio;
    cen[tid][4] = (float)cx * ratio;
  }
  __syncthreads();

  for (int it = 0; it <= KMITERS; ++it) {
    const bool last = (it == KMITERS);
    if (!last) {
      for (int i = tid; i < NSEG * 5; i += nthr) acc[i / 5][i % 5] = 0.0f;
      for (int i = tid; i < NSEG; i += nthr) cnt[i] = 0.0f;
      __syncthreads();
    }
    for (int pix = tid; pix < HW; pix += nthr) {
      int h = pix / IMG, w = pix - h * IMG;
      float f0 = img[pix], f1 = img[HW + pix], f2 = img[2 * HW + pix];
      float f3 = (float)h * ratio, f4 = (float)w * ratio;
      int best = 0;
      float bd = 3.0e38f;
      for (int k = 0; k < NSEG; ++k) {
        float d0 = f0 - cen[k][0], d1 = f1 - cen[k][1], d2 = f2 - cen[k][2];
        float d3 = f3 - cen[k][3], d4 = f4 - cen[k][4];
        float d = d0 * d0 + d1 * d1 + d2 * d2 + d3 * d3 + d4 * d4;
        if (d < bd) { bd = d; best = k; }          // ties -> first index, like argmin
      }
      if (last) {
        labels[(size_t)b * HW + pix] = best;
      } else {
        atomicAdd(&acc[best][0], f0);
        atomicAdd(&acc[best][1], f1);
        atomicAdd(&acc[best][2], f2);
        atomicAdd(&acc[best][3], f3);
        atomicAdd(&acc[best][4], f4);
        atomicAdd(&cnt[best], 1.0f);
      }
    }
    if (!last) {
      __syncthreads();
      if (tid < NSEG) {
        float c = cnt[tid];
        if (c > 0.0f) {                            // where(cnt>0, sum/cnt, old)
          float ic = 1.0f / c;
          for (int j = 0; j < 5; ++j) cen[tid][j] = acc[tid][j] * ic;
        }
      }
      __syncthreads();
    }
  }
}

// ---------------------------------------------------------------------------
__global__ void zero_kernel(float* __restrict__ p, long long n) {
  long long i = (long long)blockIdx.x * blockDim.x + threadIdx.x;
  long long stride = (long long)gridDim.x * blockDim.x;
  for (; i < n; i += stride) p[i] = 0.0f;
}

// ---------------------------------------------------------------------------
// Stage 1 (row resize): tmp[(k,p,c), w] += A[p,h] * img[c,h,w] for k=label(h,w).
// A's antialias support means each h feeds <=3 output rows p -> sparse scatter.
// ---------------------------------------------------------------------------
__global__ void scatter_kernel(const float* __restrict__ x, const int* __restrict__ labels,
                               const float* __restrict__ A, float* __restrict__ tmp) {
  int gid = blockIdx.x * blockDim.x + threadIdx.x;   // grid sized exactly B*HW
  int b = gid / HW;
  int pix = gid - b * HW;
  int h = pix / IMG, w = pix - h * IMG;
  int k = labels[(size_t)b * HW + pix];
  const float* img = x + (size_t)b * NCH * HW;
  float v0 = img[pix], v1 = img[HW + pix], v2 = img[2 * HW + pix];
  float* tb = tmp + (size_t)b * ROWS * IMG;
  int pc = h / 14;
  int plo = pc > 0 ? pc - 1 : 0;
  int phi = pc < PSZ - 1 ? pc + 1 : PSZ - 1;
  for (int p = plo; p <= phi; ++p) {
    float wgt = A[p * IMG + h];
    if (wgt != 0.0f) {
      size_t base = (size_t)((k * PSZ + p) * NCH) * IMG + w;
      atomicAdd(&tb[base], wgt * v0);
      atomicAdd(&tb[base + IMG], wgt * v1);
      atomicAdd(&tb[base + 2 * IMG], wgt * v2);
    }
  }
}

// ---------------------------------------------------------------------------
// Stage 2 (column resize): out2[r, q] = sum_w tmp[r, w] * A[q, w]
// Dense fp32 GEMM [4800 x 224] x [224 x 16] per image via V_WMMA_F32_16X16X4_F32.
// Two waves per block; 32x224 tmp tile staged into LDS by the TDM.
// ---------------------------------------------------------------------------
__global__ __launch_bounds__(64) void gemm2_kernel(const float* __restrict__ tmp,
                                                   const float* __restrict__ A,
                                                   float* __restrict__ out2) {
  __shared__ float sA[PSZ * IMG];    // resize matrix [q][w], 14 KB
  __shared__ float sT[32 * IMG];     // tmp row-tile, 28 KB
  const int b = blockIdx.x / (ROWS / 32);
  const int tile = blockIdx.x % (ROWS / 32);
  const int t0 = tile * 32;
  const int tid = threadIdx.x;

  for (int i = tid; i < PSZ * IMG; i += 64) sA[i] = A[i];

  const float* src = tmp + ((size_t)b * ROWS + t0) * IMG;
#ifdef USE_TDM
  if (tid < 32) {                    // wave 0 drives the Tensor Data Mover
    unsigned long long ga = (unsigned long long)(size_t)src;
    unsigned ldsa = (unsigned)(size_t)(&sT[0]);    // low 32 bits = LDS byte offset
    u32x4 g0;
    g0[0] = 1u;                                          // count=1, user descriptor
    g0[1] = ldsa;                                        // lds_addr
    g0[2] = (unsigned)(ga & 0xffffffffu);                // global_addr[31:0]
    g0[3] = (unsigned)((ga >> 32) & 0x01ffffffu) | (2u << 30);  // addr[56:32] | type=2
    i32x8 g1;
    g1[0] = (2 << 16);                                   // data_size = 4 bytes
    g1[1] = (IMG << 16);                                 // tensor_dim0 = 224 (lo16)
    g1[2] = (ROWS << 16);                                // dim0 hi=0 | tensor_dim1 lo16
    g1[3] = (IMG << 16);                                 // dim1 hi=0 | tile_dim0 = 224
    g1[4] = 32;                                          // tile_dim1 = 32, tile_dim2 = 0
    g1[5] = IMG;                                         // tensor_dim0_stride lo32
    g1[6] = 0;
    g1[7] = 0;
    i32x4 gz4 = (i32x4){0, 0, 0, 0};                     // 2D tensor: groups 2/3 unused
    i32x8 gz8 = (i32x8){0, 0, 0, 0, 0, 0, 0, 0};         // 6-arg form: extra group
    __builtin_amdgcn_tensor_load_to_lds(g0, g1, gz4, gz4, gz8, 0);
    __builtin_amdgcn_s_wait_tensorcnt(0);
  }
#else
  for (int i = tid; i < 32 * IMG; i += 64) sT[i] = src[i];
#endif
  __syncthreads();

  const int wave = tid >> 5;
  const int lane = tid & 31;
  const int half = lane >> 4;
  const int rc = lane & 15;          // A-frag row (M) == B-frag col (N)
  const int koff = half * 2;         // lanes 16-31 carry K+2,K+3 (ISA 16x4 layout)
  const float* ta = &sT[(wave * PSZ + rc) * IMG];
  const float* tbp = &sA[rc * IMG];

  v8f c = {};
#pragma unroll 8
  for (int kt = 0; kt < IMG; kt += 4) {
    v2f a;  a.x = ta[kt + koff];   a.y = ta[kt + koff + 1];
    v2f bb; bb.x = tbp[kt + koff]; bb.y = tbp[kt + koff + 1];
    c = __builtin_amdgcn_wmma_f32_16x16x4_f32(false, a, false, bb,
                                              (short)0, c, false, false);
  }

  float* dst = out2 + (size_t)b * SPF;
#pragma unroll
  for (int j = 0; j < 8; ++j) {      // C layout: VGPR j -> M=j (+8 for lanes 16-31)
    int m = t0 + wave * PSZ + j + 8 * half;
    dst[(size_t)m * PSZ + rc] = c[j];
  }
}

// ---------------------------------------------------------------------------
// Final permute: out[b, s, m] with sp.view(B, K*C, P, P).flatten(2).T semantics.
// flat = m*256 + s over sp's row-major [k][p][q][c] order.
// ---------------------------------------------------------------------------
__global__ void permute_kernel(const float* __restrict__ out2, float* __restrict__ out,
                               int total) {
  int idx = blockIdx.x * blockDim.x + threadIdx.x;
  if (idx >= total) return;
  int b = idx / (PSZ * PSZ * NSEG * NCH);
  int r = idx - b * (PSZ * PSZ * NSEG * NCH);
  int s = r / (NSEG * NCH);
  int m = r - s * (NSEG * NCH);
  int flat = m * (PSZ * PSZ) + s;            // raw reinterpret index into sp[k,p,q,c]
  int k = flat / (PSZ * PSZ * NCH);
  int rem = flat - k * (PSZ * PSZ * NCH);
  int p = rem / (PSZ * NCH);
  int rem2 = rem - p * (PSZ * NCH);
  int q = rem2 / NCH;
  int ch = rem2 - q * NCH;
  // out2 is laid out as [(k,p,c), q]
  out[idx] = out2[(size_t)b * SPF + (size_t)(((k * PSZ + p) * NCH + ch) * PSZ + q)];
}

// ---------------------------------------------------------------------------
extern "C" void kernel_launch(void* const* d_in, const int* in_sizes, int n_in,
                              void* d_out, int out_size, void* d_ws, size_t ws_size,
                              hipStream_t stream) {
  const float* x = (const float*)d_in[0];
  float* out = (float*)d_out;
  const int B = in_sizes[0] / (NCH * HW);    // 8

  char* ws = (char*)d_ws;
  float* A = (float*)ws;                                     // 3584 floats
  int* labels = (int*)(ws + 16384);                          // B*HW ints
  float* tmp = (float*)(ws + 16384 + (size_t)B * HW * 4);    // B*4800*224 floats
  float* out2 = tmp + (size_t)B * ROWS * IMG;                // B*76800 floats

  build_A_kernel<<<1, 32, 0, stream>>>(A);
  slic_kernel<<<B, 1024, 0, stream>>>(x, labels);
  long long ntmp = (long long)B * ROWS * IMG;
  zero_kernel<<<1024, 256, 0, stream>>>(tmp, ntmp);
  scatter_kernel<<<(B * HW) / 256, 256, 0, stream>>>(x, labels, A, tmp);
  gemm2_kernel<<<B * (ROWS / 32), 64, 0, stream>>>(tmp, A, out2);
  int total = B * PSZ * PSZ * NSEG * NCH;
  permute_kernel<<<(total + 255) / 256, 256, 0, stream>>>(out2, out, total);
}